// Img2video_50560355008514
// MI455X (gfx1250) — compile-verified
//
#include <hip/hip_runtime.h>
#include <stdint.h>

// Problem constants (match reference)
#define HH   1024
#define WW   2048
#define KK   32
#define FF   224
#define FNN  7
#define SSS  49
#define TTT  32
#define BBB  4
#define CCC  3
#define PI_F 3.14159265358979323846f

typedef uint32_t u32x4 __attribute__((ext_vector_type(4)));
typedef uint32_t u32x8 __attribute__((ext_vector_type(8)));

// Fast asin for |x| <= 1 (input pre-clamped). Cephes poly, ~1e-7 rad error:
// far below the pi/1024 index granularity of the output trunc.
__device__ __forceinline__ float fast_asinf(float x) {
    const float ax  = fabsf(x);
    const bool  big = ax > 0.5f;
    const float z   = big ? (1.0f - ax) * 0.5f : ax * ax;
    const float s   = big ? __builtin_amdgcn_sqrtf(z) : ax;
    float p = fmaf(z, 4.2163199048e-2f, 2.4181311049e-2f);
    p = fmaf(p, z, 4.5470025998e-2f);
    p = fmaf(p, z, 7.4953002686e-2f);
    p = fmaf(p, z, 1.6666752422e-1f);
    float r = fmaf(s * z, p, s);                      // s + s*z*P(z)
    if (big) r = fmaf(-2.0f, r, 0.5f * PI_F);         // pi/2 - 2r
    return copysignf(r, x);
}

// Fast atan, full range incl. +-inf (rcp(inf)=0 -> pi/2). Degree-11 odd
// minimax on [0,1] + reciprocal fold, ~2e-7 rad error.
__device__ __forceinline__ float fast_atanf(float x) {
    const float ax  = fabsf(x);
    const bool  inv = ax > 1.0f;
    const float z   = inv ? __builtin_amdgcn_rcpf(ax) : ax;
    const float z2  = z * z;
    float p = fmaf(z2, -1.1721200e-2f, 5.2653320e-2f);
    p = fmaf(p, z2, -1.1643287e-1f);
    p = fmaf(p, z2,  1.9354346e-1f);
    p = fmaf(p, z2, -3.3262347e-1f);
    p = fmaf(p, z2,  9.9997726e-1f);
    float r = z * p;
    if (inv) r = 0.5f * PI_F - r;
    return copysignf(r, x);
}

// One workgroup per (b, t, s) patch. 256 threads, 4 pixels each (32x32 tile),
// 3 channels gathered into LDS, then wave 0 fires 3 TDM tensor stores.
__global__ __launch_bounds__(256) void img2video_patch_kernel(
    const float* __restrict__ images,      // [B,C,H,W]
    const float* __restrict__ scanpaths,   // [B,S,T,2]
    const int*   __restrict__ masking,     // [B]
    float*       __restrict__ out)         // [B,C,T,F,F]
{
    __shared__ float tile[CCC][KK][KK];    // 12 KB staging tile
    __shared__ float s_kx[KK];             // kerX[e] = tan(rx[e]*delta)
    __shared__ float s_ky[KK];             // kerY[d] = tan(rx[d]*delta)/cos(rx[d]*delta)

    const int bid = blockIdx.x;
    const int s   = bid % SSS;
    const int t   = (bid / SSS) % TTT;
    const int b   = bid / (SSS * TTT);
    const int a   = s / FNN;               // fragment row
    const int fb  = s % FNN;               // fragment col

    const int tid = threadIdx.x;

    // --- Separable kernel tables: 64 transcendental evals per block ---
    if (tid < 2 * KK) {
        const int j  = tid & (KK - 1);
        const int rx = (j < 16) ? (j - 16) : (j - 15);    // arange(-16..16) \ {0}
        const float ang = (float)rx * (PI_F / (float)HH); // delta_lat == delta_lon
        if (tid < KK) s_kx[j] = tanf(ang);
        else          s_ky[j] = tanf(ang) / cosf(ang);
    }

    // Uniform (block-level) mask check; masked frames skip all image reads.
    const bool active = t < masking[b];

    float sin_lat = 0.0f, cos_lat = 0.0f, lon = 0.0f;
    const float* __restrict__ img_b = images + (size_t)b * (CCC * HH * WW);
    if (active) {
        // Block-uniform scanpath sample -> scalarized s_loads
        const int spb = (((b * SSS) + s) * TTT + t) * 2;
        const float sp_y = scanpaths[spb + 0];
        const float sp_x = scanpaths[spb + 1];
        const float yq = truncf(sp_y * (float)HH - 0.5f);
        const float xq = truncf(sp_x * (float)WW - 0.5f);
        const float lat = (yq / (float)HH - 0.5f) * PI_F;
        lon     = (xq / (float)WW - 0.5f) * (2.0f * PI_F);
        sin_lat = sinf(lat);
        cos_lat = cosf(lat);
    }

    __syncthreads();   // kernel tables ready

    if (active) {
        #pragma unroll
        for (int i = 0; i < 4; ++i) {
            const int p = tid + i * 256;        // pixel id within 32x32 tile
            const int d = p >> 5;               // kernel row (wave-uniform)
            const int e = p & 31;               // kernel col (== lane)

            const float kx = s_kx[e];           // conflict-free ds_load
            const float ky = s_ky[d];           // broadcast ds_load
            const float r2 = fmaf(kx, kx, ky * ky);
            // nu = atan(rho): cos_nu = 1/sqrt(1+rho^2); sin_nu = rho*cos_nu.
            const float cos_nu = __builtin_amdgcn_rsqf(1.0f + r2);

            // arg = cos_nu*sin_lat + ky*sin_nu*cos_lat/rho  (sin_nu/rho == cos_nu)
            float arg = fmaf(cos_nu, sin_lat, (ky * cos_nu) * cos_lat);
            arg = fminf(1.0f, fmaxf(-1.0f, arg));
            const float plat = fast_asinf(arg);

            // kx*sin_nu / (rho*cos_lat*cos_nu - ky*sin_lat*sin_nu)
            //   == kx / (cos_lat - ky*sin_lat)   (positive factor rho*cos_nu cancels)
            const float den  = fmaf(-ky, sin_lat, cos_lat);
            const float plon = fast_atanf(kx * __builtin_amdgcn_rcpf(den)) + lon;

            // li = trunc((plat/pi + 0.5)*H - 0.5) pythonmod H ; H = 2^10.
            // (plat/pi + 0.5)*1024 - 0.5 == plat*(1024/pi) + 511.5
            const float lv = fmaf(plat, (float)HH / PI_F, 511.5f);
            const int li = ((int)lv) & (HH - 1);          // cvt_i32 truncs to zero

            // plon in (-3pi/2, 3pi/2) -> uv in (-512, 2560): python float-mod
            // reduces to a single conditional +-W. Then trunc + pythonmod W.
            float uv = fmaf(plon, (float)WW / (2.0f * PI_F), (float)(WW / 2));
            uv = (uv < 0.0f) ? uv + (float)WW
                             : ((uv >= (float)WW) ? uv - (float)WW : uv);
            const int lo = ((int)(uv - 0.5f)) & (WW - 1);

            const int flat = li * WW + lo;
            #pragma unroll
            for (int c = 0; c < CCC; ++c)
                tile[c][d][e] = img_b[c * (HH * WW) + flat];
        }
    } else {
        // Zero-fill the tile for masked frames (no image reads at all)
        float* lp = &tile[0][0][0];
        #pragma unroll
        for (int i = 0; i < 4; ++i) {
            const int p = tid + i * 256;
            lp[p]        = 0.0f;
            lp[p + 1024] = 0.0f;
            lp[p + 2048] = 0.0f;
        }
    }

    __syncthreads();   // LDS tile complete before TDM reads it

    // --- Tensor Data Mover store: wave 0 only (TDM ignores EXEC; the
    // wave-uniform branch keeps the other 7 waves from issuing it). ---
    if (tid < 32) {
        // D# group 1 (shared by all 3 channels): 2D tile, 32x32 fp32,
        // row stride = F elements; tensor dims huge so no OOB drops;
        // workgroup_mask = 0 (not in a cluster).
        u32x8 g1;
        g1[0] = 0x00020000u;   // [17:16] data_size = 2 (4 bytes); mask/pad/iter = 0
        g1[1] = 0x00000000u;   // atomic_barrier_addr = 0; tensor_dim0 lo16 = 0
        g1[2] = 0x00004000u;   // tensor_dim0 = 0x40000000; tensor_dim1 lo16 = 0
        g1[3] = 0x00204000u;   // tensor_dim1 = 0x40000000; tile_dim0 = 32
        g1[4] = 0x00000020u;   // tile_dim1 = 32; tile_dim2 = 0 (2D tile)
        g1[5] = (uint32_t)FF;  // tensor_dim0_stride = 224 (low 32)
        g1[6] = 0x00000000u;   // dim0_stride hi16 = 0; dim1_stride lo16 = 0
        g1[7] = 0x00000000u;   // dim1_stride hi32 = 0 (unused for 2D)

        // Tile origin in the output tensor: out[b][c][t][a*K][fb*K]
        const uint64_t elem_off =
            ((uint64_t)(b * CCC) * TTT + (uint64_t)t) * (uint64_t)(FF * FF) +
            (uint64_t)(a * KK) * FF + (uint64_t)(fb * KK);
        const uint64_t out_base = (uint64_t)(uintptr_t)out + 4ull * elem_off;
        const uint32_t lds_base = (uint32_t)(uintptr_t)(&tile[0][0][0]);

        #pragma unroll
        for (int c = 0; c < CCC; ++c) {
            const uint64_t ga = out_base + 4ull * (uint64_t)c * (TTT * FF * FF);
            u32x4 g0;
            g0[0] = 0x00000001u;                        // count = 1, user mode
            g0[1] = lds_base + (uint32_t)c * (KK * KK * 4u);   // lds_addr
            g0[2] = (uint32_t)ga;                       // global_addr[31:0]
            g0[3] = (uint32_t)(ga >> 32) | 0x80000000u; // global_addr[56:32] | type=2
            asm volatile("tensor_store_from_lds %0, %1"
                         :: "s"(g0), "s"(g1)
                         : "memory");
        }
        // Keep LDS alive until the DMA drains.
        __builtin_amdgcn_s_wait_tensorcnt(0);
    }
}

extern "C" void kernel_launch(void* const* d_in, const int* in_sizes, int n_in,
                              void* d_out, int out_size, void* d_ws, size_t ws_size,
                              hipStream_t stream) {
    (void)in_sizes; (void)n_in; (void)out_size; (void)d_ws; (void)ws_size;
    const float* images    = (const float*)d_in[0];
    const float* scanpaths = (const float*)d_in[1];
    const int*   masking   = (const int*)d_in[2];
    float*       out       = (float*)d_out;

    // One block per (b, t, s) patch
    dim3 grid(BBB * TTT * SSS);   // 4 * 32 * 49 = 6272
    dim3 block(256);
    img2video_patch_kernel<<<grid, block, 0, stream>>>(images, scanpaths, masking, out);
}